// FocalLoss_40707700032510
// MI455X (gfx1250) — compile-verified
//
#include <hip/hip_runtime.h>
#include <math.h>

// ---------------------------------------------------------------------------
// RetinaNet focal loss, fused single pass, MI455X (gfx1250, wave32).
// Memory-bound: 256MB classification stream -> ~12us floor @ 23.3 TB/s.
// ---------------------------------------------------------------------------

typedef __attribute__((ext_vector_type(2))) float v2f;
typedef __attribute__((ext_vector_type(8))) float v8f;

#define FL_ALPHA 0.25f
#define FL_CLIP  1e-4f

// Exact wave32 sum using V_WMMA_F32_16X16X4_F32 with B = ones.
// A layout (32-bit A 16x4): lanes 0-15 hold (M=lane, K=0/1) in vgpr0/1,
// lanes 16-31 hold (M=lane-16, K=2/3). With a.vgpr0 = p(lane) and
// a.vgpr1 = shfl_xor(p,8):
//   D[m][n] = p(m) + p(m^8) + p(m+16) + p((m^8)+16)   (exact: *1.0 products)
// Rows 0..7 cover the 8 disjoint groups of 4 lanes, and every lane holds
// D rows 0..7 (or their duplicates 8..15) in its 8 dest VGPRs, so
// d[0]+...+d[7] == sum over all 32 lanes, in every lane. EXEC must be ~0.
__device__ __forceinline__ float wave_sum32(float v) {
    v2f a;
    a.x = v;
    a.y = __shfl_xor(v, 8, 32);
    v2f ones;
    ones.x = 1.0f;
    ones.y = 1.0f;
    v8f c = {0.f, 0.f, 0.f, 0.f, 0.f, 0.f, 0.f, 0.f};
    v8f d = __builtin_amdgcn_wmma_f32_16x16x4_f32(
        /*neg_a=*/false, a, /*neg_b=*/false, ones,
        /*c_mod=*/(short)0, c, /*reuse_a=*/false, /*reuse_b=*/false);
    return ((d[0] + d[1]) + (d[2] + d[3])) + ((d[4] + d[5]) + (d[6] + d[7]));
}

__global__ void zero_acc_kernel(float* acc, int n) {
    int i = blockIdx.x * blockDim.x + threadIdx.x;
    if (i < n) acc[i] = 0.0f;
}

__global__ void __launch_bounds__(256)
focal_main_kernel(const float* __restrict__ cls,   // [B,A,C]
                  const float* __restrict__ reg,   // [B,A,4]
                  const float* __restrict__ anc,   // [A,4]
                  const float* __restrict__ ann,   // [B,M,5]
                  float* __restrict__ acc,         // [3*B] cls_sum / reg_sum / pos_cnt
                  int A, int C, int M, int B)
{
    const int b   = blockIdx.y;
    const int tid = threadIdx.x;
    const int a   = blockIdx.x * blockDim.x + tid;
    const bool active = (a < A);

    __shared__ float s_ann[64 * 5];
    __shared__ float s_red[3 * 8];

    const int m_use = (M > 64) ? 64 : M;
    for (int i = tid; i < m_use * 5; i += blockDim.x)
        s_ann[i] = ann[(size_t)b * M * 5 + i];
    __syncthreads();

    float cls_part = 0.0f, reg_part = 0.0f, pos_part = 0.0f;

    if (active) {
        const size_t row = (size_t)b * A + a;
        const float* crow = cls + row * (size_t)C;
        // Start pulling the 320B class row while IoU math runs.
        __builtin_prefetch(crow, 0, 0);
        __builtin_prefetch(crow + 64, 0, 0);

        const float4 ab = *(const float4*)(anc + (size_t)a * 4);
        const float aw = ab.z - ab.x, ah = ab.w - ab.y;
        const float acx = ab.x + 0.5f * aw, acy = ab.y + 0.5f * ah;
        const float aarea = aw * ah;

        // IoU argmax over M boxes (first-max semantics like jnp.argmax).
        float best = -INFINITY;
        int bestj = 0;
        for (int j = 0; j < m_use; ++j) {
            const float bx1 = s_ann[j * 5 + 0], by1 = s_ann[j * 5 + 1];
            const float bx2 = s_ann[j * 5 + 2], by2 = s_ann[j * 5 + 3];
            const float lab = s_ann[j * 5 + 4];
            float iw = fminf(ab.z, bx2) - fmaxf(ab.x, bx1);
            float ih = fminf(ab.w, by2) - fmaxf(ab.y, by1);
            iw = fmaxf(iw, 0.0f);
            ih = fmaxf(ih, 0.0f);
            const float inter = iw * ih;
            const float ua = fmaxf(aarea + (bx2 - bx1) * (by2 - by1) - inter, 1e-8f);
            float iou = inter / ua;
            iou = (lab != -1.0f) ? iou : -1.0f;
            if (iou > best) { best = iou; bestj = j; }
        }
        const bool pos = (best >= 0.5f);
        const bool neg = (best < 0.4f);
        const int  label = (int)s_ann[bestj * 5 + 4];

        // Focal classification loss: only pos/neg anchors contribute
        // (targets == -1 rows are exactly zero, so skip the 320B read).
        if (pos | neg) {
            const float4* cp = (const float4*)crow;
            const int n4 = C >> 2;
            for (int i = 0; i < n4; ++i) {
                const float4 p4 = cp[i];
                #pragma unroll
                for (int k = 0; k < 4; ++k) {
                    const int ci = i * 4 + k;
                    float q = (k == 0) ? p4.x : (k == 1) ? p4.y : (k == 2) ? p4.z : p4.w;
                    q = fminf(fmaxf(q, FL_CLIP), 1.0f - FL_CLIP);
                    const bool hit = pos && (ci == label);          // target == 1
                    const float af   = hit ? FL_ALPHA : (1.0f - FL_ALPHA);
                    const float fw   = hit ? (1.0f - q) : q;        // focal weight base
                    const float larg = hit ? q : (1.0f - q);        // bce log argument
                    cls_part += af * fw * fw * (-__logf(larg));     // gamma == 2
                }
            }
        }

        // Smooth-L1 regression loss: positives only (rare -> cheap reads).
        if (pos) {
            pos_part = 1.0f;
            const float g0 = s_ann[bestj * 5 + 0], g1 = s_ann[bestj * 5 + 1];
            const float g2 = s_ann[bestj * 5 + 2], g3 = s_ann[bestj * 5 + 3];
            float gw = g2 - g0, gh = g3 - g1;
            const float gcx = g0 + 0.5f * gw, gcy = g1 + 0.5f * gh;
            gw = fmaxf(gw, 1.0f);
            gh = fmaxf(gh, 1.0f);
            // BOX_STD = [0.1,0.1,0.2,0.2] -> multiply by 10,10,5,5
            const float t0 = ((gcx - acx) / aw) * 10.0f;
            const float t1 = ((gcy - acy) / ah) * 10.0f;
            const float t2 = __logf(gw / aw) * 5.0f;
            const float t3 = __logf(gh / ah) * 5.0f;
            const float4 r4 = *(const float4*)(reg + row * 4);
            float s = 0.0f;
            {
                float d0 = fabsf(t0 - r4.x);
                s += (d0 <= (1.0f / 9.0f)) ? 4.5f * d0 * d0 : (d0 - (0.5f / 9.0f));
                float d1 = fabsf(t1 - r4.y);
                s += (d1 <= (1.0f / 9.0f)) ? 4.5f * d1 * d1 : (d1 - (0.5f / 9.0f));
                float d2 = fabsf(t2 - r4.z);
                s += (d2 <= (1.0f / 9.0f)) ? 4.5f * d2 * d2 : (d2 - (0.5f / 9.0f));
                float d3 = fabsf(t3 - r4.w);
                s += (d3 <= (1.0f / 9.0f)) ? 4.5f * d3 * d3 : (d3 - (0.5f / 9.0f));
            }
            reg_part = s;
        }
    }

    // ---- reduction: wave32 via WMMA (EXEC all-ones here: fully reconverged,
    // no early returns anywhere above), then cross-wave via LDS. ----
    const int wid  = tid >> 5;
    const int lane = tid & 31;
    const float cs = wave_sum32(cls_part);
    const float rs = wave_sum32(reg_part);
    const float ps = wave_sum32(pos_part);
    if (lane == 0) {
        s_red[wid]      = cs;
        s_red[8 + wid]  = rs;
        s_red[16 + wid] = ps;
    }
    __syncthreads();
    if (tid == 0) {
        float c = 0.f, r = 0.f, p = 0.f;
        const int nw = blockDim.x >> 5;
        for (int w = 0; w < nw; ++w) {
            c += s_red[w];
            r += s_red[8 + w];
            p += s_red[16 + w];
        }
        atomicAdd(&acc[b], c);
        atomicAdd(&acc[B + b], r);
        atomicAdd(&acc[2 * B + b], p);
    }
}

__global__ void finalize_kernel(const float* __restrict__ acc,
                                float* __restrict__ out, int B) {
    if (blockIdx.x == 0 && threadIdx.x == 0) {
        float cm = 0.0f, rm = 0.0f;
        for (int b = 0; b < B; ++b) {
            const float np = acc[2 * B + b];
            const float d = fmaxf(np, 1.0f);
            cm += acc[b] / d;
            rm += (np > 0.0f) ? (acc[B + b] / (4.0f * d)) : 0.0f;
        }
        out[0] = cm / (float)B;
        out[1] = rm / (float)B;
    }
}

extern "C" void kernel_launch(void* const* d_in, const int* in_sizes, int n_in,
                              void* d_out, int out_size, void* d_ws, size_t ws_size,
                              hipStream_t stream) {
    const float* cls = (const float*)d_in[0];   // [B,A,C]
    const float* reg = (const float*)d_in[1];   // [B,A,4]
    const float* anc = (const float*)d_in[2];   // [1,A,4]
    const float* ann = (const float*)d_in[3];   // [B,M,5]

    const int A = in_sizes[2] / 4;
    const int B = (int)((long long)in_sizes[1] / ((long long)A * 4));
    const int C = (int)((long long)in_sizes[0] / ((long long)B * A));
    const int M = (int)((long long)in_sizes[3] / ((long long)B * 5));

    float* acc = (float*)d_ws;                  // [3*B] accumulators

    zero_acc_kernel<<<1, 64, 0, stream>>>(acc, 3 * B);

    dim3 grid((A + 255) / 256, B);
    focal_main_kernel<<<grid, 256, 0, stream>>>(cls, reg, anc, ann, acc, A, C, M, B);

    finalize_kernel<<<1, 32, 0, stream>>>(acc, (float*)d_out, B);
}